// GNNAgent_38852274159906
// MI455X (gfx1250) — compile-verified
//
#include <hip/hip_runtime.h>

#define N_KW  65536
#define N_DOC 65536
#define DM    128

typedef __attribute__((ext_vector_type(2))) float v2f;
typedef __attribute__((ext_vector_type(8))) float v8f;

// ---------------------------------------------------------------------------
// WMMA GEMM: out[M x 128] = act(A[M x K] @ W[K x 128] + bias)
// 256 threads = 8 wave32; each block computes 128 rows; each wave a 16x128
// strip (8 accumulator tiles). W staged in LDS. K in {8,16,128}.
// Fragment layouts per CDNA5 ISA 7.12.2:
//   A 16x4 f32 : lane L -> M = L%16, VGPR{0,1} = K{2*(L/16), 2*(L/16)+1}
//   B 4x16 f32 : lane L -> N = L%16, VGPR{0,1} = K{2*(L/16), 2*(L/16)+1}
//   C/D 16x16  : lane L -> N = L%16, VGPR v -> M = v + 8*(L/16)
// ---------------------------------------------------------------------------
template <int K, bool BIAS, bool RELU>
__global__ __launch_bounds__(256) void gemm_wmma(const float* __restrict__ A,
                                                 const float* __restrict__ W,
                                                 const float* __restrict__ bias,
                                                 float* __restrict__ out) {
  __shared__ float sW[K * DM];
  for (int i = threadIdx.x * 4; i < K * DM; i += 256 * 4) {
    *reinterpret_cast<float4*>(&sW[i]) = *reinterpret_cast<const float4*>(&W[i]);
  }
  __syncthreads();

  const int lane = threadIdx.x & 31;
  const int wave = threadIdx.x >> 5;
  const int m    = lane & 15;   // M (for A) / N (for B,C,D) sub-index
  const int kh   = lane >> 4;   // which K-pair of the 4-wide K slab
  const int rowBase = blockIdx.x * 128 + wave * 16;
  const float* __restrict__ arow = A + (size_t)(rowBase + m) * K + kh * 2;

  v8f acc[8] = {};

  for (int k0 = 0; k0 < K; k0 += 4) {
    const v2f a = *reinterpret_cast<const v2f*>(arow + k0);
    const float* __restrict__ wp = &sW[(k0 + kh * 2) * DM + m];
#pragma unroll
    for (int nt = 0; nt < 8; ++nt) {
      v2f b;
      b.x = wp[nt * 16];
      b.y = wp[nt * 16 + DM];
      acc[nt] = __builtin_amdgcn_wmma_f32_16x16x4_f32(
          false, a, false, b, (short)0, acc[nt], false, false);
    }
  }

#pragma unroll
  for (int nt = 0; nt < 8; ++nt) {
    const int n  = nt * 16 + m;
    const float bv = BIAS ? bias[n] : 0.0f;
#pragma unroll
    for (int v = 0; v < 8; ++v) {
      float r = acc[nt][v] + bv;
      if (RELU) r = fmaxf(r, 0.0f);
      out[(size_t)(rowBase + kh * 8 + v) * DM + n] = r;
    }
  }
}

// ---------------------------------------------------------------------------
// Degree counting + normalization tables
// ---------------------------------------------------------------------------
__global__ void zero_u32(unsigned* __restrict__ p, int n) {
  int i = blockIdx.x * 256 + threadIdx.x;
  if (i < n) p[i] = 0u;
}

__global__ void zero_f32(float* __restrict__ p) {
  int i = blockIdx.x * 256 + threadIdx.x;  // one float4 per thread
  *reinterpret_cast<float4*>(p + (size_t)i * 4) = make_float4(0.f, 0.f, 0.f, 0.f);
}

__global__ void count_deg(const int* __restrict__ src, const int* __restrict__ dst,
                          unsigned* __restrict__ cnt_kw, unsigned* __restrict__ cnt_doc,
                          int nE) {
  int e = blockIdx.x * 256 + threadIdx.x;
  if (e < nE) {
    atomicAdd(&cnt_kw[src[e]], 1u);
    atomicAdd(&cnt_doc[dst[e] - N_KW], 1u);
  }
}

__global__ void make_norm(const unsigned* __restrict__ cnt_doc,
                          const unsigned* __restrict__ cnt_kw,
                          float* __restrict__ dis1_doc, float* __restrict__ dis2_kw,
                          float* __restrict__ invdeg2_kw) {
  int i = blockIdx.x * 256 + threadIdx.x;  // 0..65535
  float d1 = (float)(cnt_doc[i] + 1u);     // doc in-degree + self loop (layer 1 dst)
  float d2 = (float)(cnt_kw[i] + 1u);      // kw  in-degree + self loop (layer 2 dst)
  dis1_doc[i]   = rsqrtf(d1);
  dis2_kw[i]    = rsqrtf(d2);
  invdeg2_kw[i] = 1.0f / d2;
}

// ---------------------------------------------------------------------------
// Edge scatter-add: one wave per edge, lane l moves cols [4l, 4l+4)
// agg[dst] += H[src]  (normalization folded in later since it's per-dst)
// ---------------------------------------------------------------------------
__global__ __launch_bounds__(256) void scatter_add(const float* __restrict__ H,
                                                   const int* __restrict__ src_ids,
                                                   const int* __restrict__ dst_ids,
                                                   int src_off, int dst_off,
                                                   float* __restrict__ agg, int nE) {
  int e = blockIdx.x * 8 + (threadIdx.x >> 5);
  if (e >= nE) return;
  int lane = threadIdx.x & 31;
  int s = src_ids[e] - src_off;
  int d = dst_ids[e] - dst_off;
  const float4 v = *reinterpret_cast<const float4*>(H + (size_t)s * DM + lane * 4);
  float* p = agg + (size_t)d * DM + lane * 4;
  unsafeAtomicAdd(p + 0, v.x);
  unsafeAtomicAdd(p + 1, v.y);
  unsafeAtomicAdd(p + 2, v.z);
  unsafeAtomicAdd(p + 3, v.w);
}

// ---------------------------------------------------------------------------
// Elementwise epilogues (float4 over [65536 x 128])
// ---------------------------------------------------------------------------
__global__ void doc_post(float* __restrict__ x /* agg_doc -> x_doc in place */,
                         const float* __restrict__ dis1, const float* __restrict__ bg1) {
  int i4 = blockIdx.x * 256 + threadIdx.x;
  int base = i4 * 4;
  int d = base >> 7, c = base & 127;
  float s = dis1[d];
  float4 v = *reinterpret_cast<float4*>(x + base);
  float4 b = *reinterpret_cast<const float4*>(bg1 + c);
  v.x = fmaxf(s * v.x + b.x, 0.f);
  v.y = fmaxf(s * v.y + b.y, 0.f);
  v.z = fmaxf(s * v.z + b.z, 0.f);
  v.w = fmaxf(s * v.w + b.w, 0.f);
  *reinterpret_cast<float4*>(x + base) = v;
}

__global__ void kw_out(const float* __restrict__ agg_kw, const float* __restrict__ h2kw,
                       const float* __restrict__ dis2, const float* __restrict__ invd2,
                       const float* __restrict__ bg2, float* __restrict__ out) {
  int i4 = blockIdx.x * 256 + threadIdx.x;
  int base = i4 * 4;
  int k = base >> 7, c = base & 127;
  float s = dis2[k], t = invd2[k];
  float4 a = *reinterpret_cast<const float4*>(agg_kw + base);
  float4 h = *reinterpret_cast<const float4*>(h2kw + base);
  float4 b = *reinterpret_cast<const float4*>(bg2 + c);
  float4 r;
  r.x = fmaxf(s * a.x + t * h.x + b.x, 0.f);
  r.y = fmaxf(s * a.y + t * h.y + b.y, 0.f);
  r.z = fmaxf(s * a.z + t * h.z + b.z, 0.f);
  r.w = fmaxf(s * a.w + t * h.w + b.w, 0.f);
  *reinterpret_cast<float4*>(out + base) = r;
}

__global__ void mul_ew(float* __restrict__ a, const float* __restrict__ b) {
  int i4 = blockIdx.x * 256 + threadIdx.x;
  int base = i4 * 4;
  float4 x = *reinterpret_cast<float4*>(a + base);
  float4 y = *reinterpret_cast<const float4*>(b + base);
  x.x *= y.x; x.y *= y.y; x.z *= y.z; x.w *= y.w;
  *reinterpret_cast<float4*>(a + base) = x;
}

__global__ void doc_out(const float* __restrict__ xdoc, const float* __restrict__ xft,
                        float* __restrict__ out) {
  int i4 = blockIdx.x * 256 + threadIdx.x;
  int base = i4 * 4;
  float4 x = *reinterpret_cast<const float4*>(xdoc + base);
  float4 y = *reinterpret_cast<const float4*>(xft + base);
  x.x *= y.x; x.y *= y.y; x.z *= y.z; x.w *= y.w;
  *reinterpret_cast<float4*>(out + base) = x;
}

// ---------------------------------------------------------------------------
extern "C" void kernel_launch(void* const* d_in, const int* in_sizes, int n_in,
                              void* d_out, int out_size, void* d_ws, size_t ws_size,
                              hipStream_t stream) {
  const float* X_nodes = (const float*)d_in[0];
  const float* X_fb    = (const float*)d_in[1];
  const float* X_time  = (const float*)d_in[2];
  const int*   ek      = (const int*)d_in[3];   // [2,E]: row0 = kw src, row1 = doc dst
  const int E = in_sizes[3] / 2;
  const float* W0  = (const float*)d_in[6];
  const float* b0  = (const float*)d_in[7];
  const float* Wg1 = (const float*)d_in[8];
  const float* bg1 = (const float*)d_in[9];
  const float* Wg2 = (const float*)d_in[10];
  const float* bg2 = (const float*)d_in[11];
  const float* Wf1 = (const float*)d_in[12];
  const float* bf1 = (const float*)d_in[13];
  const float* Wf2 = (const float*)d_in[14];
  const float* bf2 = (const float*)d_in[15];
  const float* Wt1 = (const float*)d_in[16];
  const float* bt1 = (const float*)d_in[17];
  const float* Wt2 = (const float*)d_in[18];
  const float* bt2 = (const float*)d_in[19];
  float* out = (float*)d_out;

  char* w = (char*)d_ws;
  unsigned* cnt_doc = (unsigned*)w; w += (size_t)N_DOC * 4;
  unsigned* cnt_kw  = (unsigned*)w; w += (size_t)N_KW * 4;
  float* dis1  = (float*)w;         w += (size_t)N_DOC * 4;
  float* dis2  = (float*)w;         w += (size_t)N_KW * 4;
  float* invd2 = (float*)w;         w += (size_t)N_KW * 4;

  const size_t MAT = (size_t)N_KW * DM;  // 8,388,608 elements per row-buffer
  float* bufA = (float*)w; w += MAT * 4; // h0 ........ later agg_kw
  float* bufB = (float*)w; w += MAT * 4; // MLP hidden / h1 / h2doc
  float* bufC = (float*)w; w += MAT * 4; // agg_doc -> x_doc
  float* bufE = (float*)w; w += MAT * 4; // xf -> xf*xt
  float* bufG = (float*)w; w += MAT * 4; // xt tmp .... later h2kw

  const dim3 B256(256);
  const int GB   = N_KW / 128;           // 512 GEMM blocks (M = 65536)
  const int EW4  = (int)(MAT / 1024);    // float4 elementwise grids (8192)

  // 1) degrees + normalization
  zero_u32<<<(2 * N_DOC) / 256, B256, 0, stream>>>(cnt_doc, 2 * N_DOC);
  count_deg<<<(E + 255) / 256, B256, 0, stream>>>(ek, ek + E, cnt_kw, cnt_doc, E);
  make_norm<<<N_KW / 256, B256, 0, stream>>>(cnt_doc, cnt_kw, dis1, dis2, invd2);

  // 2) feedback/time MLPs, fused product into bufE
  gemm_wmma<16, true, true ><<<GB, B256, 0, stream>>>(X_fb,  Wf1, bf1, bufB);
  gemm_wmma<128, true, false><<<GB, B256, 0, stream>>>(bufB, Wf2, bf2, bufE);
  gemm_wmma<8,  true, true ><<<GB, B256, 0, stream>>>(X_time, Wt1, bt1, bufB);
  gemm_wmma<128, true, false><<<GB, B256, 0, stream>>>(bufB, Wt2, bt2, bufG);
  mul_ew<<<EW4, B256, 0, stream>>>(bufE, bufG);  // bufE = xf * xt

  // 3) layer 0 + layer 1 (kw -> doc)
  gemm_wmma<128, true,  true ><<<GB, B256, 0, stream>>>(X_nodes, W0,  b0,      bufA); // h0
  gemm_wmma<128, false, false><<<GB, B256, 0, stream>>>(bufA,    Wg1, nullptr, bufB); // h1
  zero_f32<<<EW4, B256, 0, stream>>>(bufC);
  scatter_add<<<E / 8, B256, 0, stream>>>(bufB, ek, ek + E, 0, N_KW, bufC, E);
  doc_post<<<EW4, B256, 0, stream>>>(bufC, dis1, bg1);                                // x_doc

  // 4) layer 2 (doc -> kw) with kw self-loop term
  gemm_wmma<128, false, false><<<GB, B256, 0, stream>>>(bufC, Wg2, nullptr, bufB);    // h2doc
  gemm_wmma<128, false, false><<<GB, B256, 0, stream>>>(bufA, Wg2, nullptr, bufG);    // h2kw
  zero_f32<<<EW4, B256, 0, stream>>>(bufA);                                           // agg_kw
  scatter_add<<<E / 8, B256, 0, stream>>>(bufB, ek + E, ek, N_KW, 0, bufA, E);
  kw_out<<<EW4, B256, 0, stream>>>(bufA, bufG, dis2, invd2, bg2, out);

  // 5) doc rows: x_doc * xf * xt
  doc_out<<<EW4, B256, 0, stream>>>(bufC, bufE, out + MAT);
}